// ExponentialMovingAverage_90477781058053
// MI455X (gfx1250) — compile-verified
//
#include <hip/hip_runtime.h>
#include <cstdint>
#include <cstddef>

// Problem constants (from reference): x is (B=32, C=128, T=16384) fp32, row-major in T.
#define B_DIM 32
#define C_DIM 128
#define T_DIM 16384
#define NTHREADS 256
#define K_PER_THREAD 8
#define TILE (NTHREADS * K_PER_THREAD)   // 2048 floats = 8 KB per tile
#define NTILES (T_DIM / TILE)            // 8
#define NWAVES (NTHREADS / 32)           // 8 wave32s

// ---- CDNA5 async global->LDS path (ASYNCcnt-tracked) -----------------------
#if defined(__AMDGCN__) && __has_builtin(__builtin_amdgcn_global_load_async_to_lds_b128)
#define USE_ASYNC 1
#else
#define USE_ASYNC 0
#endif

#if USE_ASYNC
typedef int v4i_t __attribute__((ext_vector_type(4)));
typedef __attribute__((address_space(1))) v4i_t* g_v4i_ptr;
typedef __attribute__((address_space(3))) v4i_t* l_v4i_ptr;

__device__ __forceinline__ void async_load_b128(const void* gsrc, void* ldst) {
  // global_load_async_to_lds_b128: lane loads 16B from global, DMA into LDS.
  __builtin_amdgcn_global_load_async_to_lds_b128(
      (g_v4i_ptr)gsrc, (l_v4i_ptr)ldst, /*offset=*/0, /*cpol=*/0);
}
#endif

template <int N>
__device__ __forceinline__ void wait_asynccnt() {
#if USE_ASYNC
#if __has_builtin(__builtin_amdgcn_s_wait_asynccnt)
  __builtin_amdgcn_s_wait_asynccnt(N);
#else
  asm volatile("s_wait_asynccnt %0" ::"i"(N) : "memory");
#endif
#endif
}

// Streaming (never re-read) output: non-temporal b128 stores keep the 192MB L2
// available for the inbound async stream instead of dirty output lines.
// Must use a native clang vector type (not HIP_vector_type) for the builtin.
typedef float v4f_t __attribute__((ext_vector_type(4)));

__device__ __forceinline__ void store_nt_f4(float* p, v4f_t v) {
  __builtin_nontemporal_store(v, (v4f_t*)p);
}

// ---------------------------------------------------------------------------
// One block per (b,c) row. Affine-scan parallelization:
//   state transform per element:  s' = w*x + a*s
//   per-thread chunk of 8  ->  (A,B) = (a^8, local_scan_end)
//   wave scan via shfl, cross-wave fold via LDS, carry chained across tiles.
// ---------------------------------------------------------------------------
__global__ __launch_bounds__(NTHREADS)
void ema_scan_kernel(const float* __restrict__ x,
                     const float* __restrict__ weights,
                     float* __restrict__ y) {
  const int row  = blockIdx.x;          // row = b*128 + c
  const int c    = row & (C_DIM - 1);
  const int tid  = threadIdx.x;
  const int lane = tid & 31;
  const int wave = tid >> 5;

  __shared__ float tileBuf[2][TILE];    // 16 KB double buffer
  __shared__ float Aw[NWAVES];
  __shared__ float Bw[NWAVES];

  const float* xrow = x + (size_t)row * T_DIM;
  float*       yrow = y + (size_t)row * T_DIM;

  float w = weights[c];
  w = fminf(fmaxf(w, 0.02f), 1.0f);
  const float a = 1.0f - w;

  // a^1 .. a^8
  float ap[K_PER_THREAD];
  ap[0] = a;
#pragma unroll
  for (int j = 1; j < K_PER_THREAD; ++j) ap[j] = ap[j - 1] * a;
  const float a8 = ap[K_PER_THREAD - 1];

  // init carry = x[row,0]; uniform recurrence then yields y0 = x0 exactly.
  float carry = xrow[0];

  // ---- prefetch tile 0 ----
  {
    const float* g = xrow;                       // tile 0 base
#if USE_ASYNC
    async_load_b128(g + tid * 4,               &tileBuf[0][tid * 4]);
    async_load_b128(g + 1024 + tid * 4,        &tileBuf[0][1024 + tid * 4]);
#else
    *(v4f_t*)&tileBuf[0][tid * 4]        = *(const v4f_t*)(g + tid * 4);
    *(v4f_t*)&tileBuf[0][1024 + tid * 4] = *(const v4f_t*)(g + 1024 + tid * 4);
#endif
  }

  for (int t = 0; t < NTILES; ++t) {
    const int buf = t & 1;

    // ---- prefetch tile t+1 into the other buffer ----
    if (t + 1 < NTILES) {
      const float* g = xrow + (t + 1) * TILE;
#if USE_ASYNC
      async_load_b128(g + tid * 4,            &tileBuf[buf ^ 1][tid * 4]);
      async_load_b128(g + 1024 + tid * 4,     &tileBuf[buf ^ 1][1024 + tid * 4]);
#else
      *(v4f_t*)&tileBuf[buf ^ 1][tid * 4]        = *(const v4f_t*)(g + tid * 4);
      *(v4f_t*)&tileBuf[buf ^ 1][1024 + tid * 4] = *(const v4f_t*)(g + 1024 + tid * 4);
#endif
    }

    // current tile's 2 async loads are the oldest outstanding (in-order done)
    if (t + 1 < NTILES) wait_asynccnt<2>();
    else                wait_asynccnt<0>();
    __syncthreads();   // all waves' LDS slices visible

    // ---- read my 8 contiguous elements from LDS ----
    float xv[K_PER_THREAD];
    {
      v4f_t u0 = *(const v4f_t*)&tileBuf[buf][tid * 8];
      v4f_t u1 = *(const v4f_t*)&tileBuf[buf][tid * 8 + 4];
      xv[0] = u0.x; xv[1] = u0.y; xv[2] = u0.z; xv[3] = u0.w;
      xv[4] = u1.x; xv[5] = u1.y; xv[6] = u1.z; xv[7] = u1.w;
    }

    // ---- local serial scan with incoming state 0 ----
    float yl[K_PER_THREAD];
    {
      float s = 0.0f;
#pragma unroll
      for (int j = 0; j < K_PER_THREAD; ++j) {
        s = fmaf(a, s, w * xv[j]);   // s' = w*x + a*s
        yl[j] = s;
      }
    }

    // ---- wave-level inclusive scan of affine pairs (A,B) ----
    float A = a8, Bv = yl[K_PER_THREAD - 1];
#pragma unroll
    for (int d = 1; d < 32; d <<= 1) {
      float Apv = __shfl_up(A, d, 32);
      float Bpv = __shfl_up(Bv, d, 32);
      if (lane >= d) { Bv = fmaf(A, Bpv, Bv); A = A * Apv; }
    }
    // exclusive within wave
    float A_ex = __shfl_up(A, 1, 32);
    float B_ex = __shfl_up(Bv, 1, 32);
    if (lane == 0) { A_ex = 1.0f; B_ex = 0.0f; }

    if (lane == 31) { Aw[wave] = A; Bw[wave] = Bv; }
    __syncthreads();

    // ---- cross-wave prefix (serial fold over <=8 entries) ----
    float WpA = 1.0f, WpB = 0.0f;
    for (int i = 0; i < wave; ++i) { WpB = fmaf(Aw[i], WpB, Bw[i]); WpA *= Aw[i]; }
    const float s_wave_in = fmaf(WpA, carry, WpB);
    const float s_in      = fmaf(A_ex, s_wave_in, B_ex);

    // ---- fix-up and store: y[j] = yl[j] + a^(j+1) * s_in ----
    {
      v4f_t o0, o1;
      o0.x = fmaf(ap[0], s_in, yl[0]);
      o0.y = fmaf(ap[1], s_in, yl[1]);
      o0.z = fmaf(ap[2], s_in, yl[2]);
      o0.w = fmaf(ap[3], s_in, yl[3]);
      o1.x = fmaf(ap[4], s_in, yl[4]);
      o1.y = fmaf(ap[5], s_in, yl[5]);
      o1.z = fmaf(ap[6], s_in, yl[6]);
      o1.w = fmaf(ap[7], s_in, yl[7]);
      float* g = yrow + t * TILE + tid * 8;
      store_nt_f4(g,     o0);
      store_nt_f4(g + 4, o1);
    }

    // ---- advance carry across the whole tile (uniform) ----
#pragma unroll
    for (int i = 0; i < NWAVES; ++i) carry = fmaf(Aw[i], carry, Bw[i]);

    __syncthreads();   // protect Aw/Bw and tile buffer reuse next iteration
  }
}

extern "C" void kernel_launch(void* const* d_in, const int* in_sizes, int n_in,
                              void* d_out, int out_size, void* d_ws, size_t ws_size,
                              hipStream_t stream) {
  const float* x       = (const float*)d_in[0];   // (32,128,16384) fp32
  const float* weights = (const float*)d_in[1];   // (128,) fp32
  float*       out     = (float*)d_out;           // (32,128,16384) fp32
  (void)in_sizes; (void)n_in; (void)out_size; (void)d_ws; (void)ws_size;

  ema_scan_kernel<<<dim3(B_DIM * C_DIM), dim3(NTHREADS), 0, stream>>>(x, weights, out);
}